// HPUAttentionImpl_28321014350112
// MI455X (gfx1250) — compile-verified
//
#include <hip/hip_runtime.h>
#include <math.h>

// Problem constants (match reference)
#define B_    64
#define H_    32
#define KVH_  8
#define D_    128
#define BS_   128
#define G_    4          // H_ / KVH_
#define NB_   1024       // B_ * BLOCKS_PER_SEQ(16)
#define SCALE_ 0.08838834764831845f   // 1/sqrt(128)

#define LDSTR 132        // LDS row stride (floats): conflict-free + 8B aligned rows

typedef float v2f __attribute__((ext_vector_type(2)));
typedef float v8f __attribute__((ext_vector_type(8)));

// ---------------------------------------------------------------------------
// Phase 1: one workgroup per (block n, kv-head k).
//   S[g, s]  = (SCALE*q[g,:]) . K[s,:]  + bias[n,s]      (s = 0..127)
//   m[g]     = max_s S[g,s]
//   P[g, s]  = exp(S[g,s] - m[g]);   l[g] = sum_s P[g,s]
//   O[g, d]  = sum_s P[g,s] * V[s, d]
// The appended token (k_new/v_new) is overlaid at load time (pointer select)
// so the input caches are never mutated.
// 256 threads = 8 waves; wave w owns token tile [16w,16w+16) for scores and
// d-column tile [16w,16w+16) for PV. All matmuls: V_WMMA_F32_16X16X4_F32.
// A-operand LDS arrays are padded to 16 rows (rows 4..15 = 0) so fragment
// loads are unconditional ds_load_b64 (no exec-masked predication).
// ---------------------------------------------------------------------------
__global__ __launch_bounds__(256) void pa_phase1(
    const float* __restrict__ query, const float* __restrict__ knew,
    const float* __restrict__ vnew,  const float* __restrict__ kcache,
    const float* __restrict__ vcache, const int* __restrict__ block_list,
    const int* __restrict__ block_groups, const float* __restrict__ block_bias,
    const int* __restrict__ block_indices, const int* __restrict__ block_offsets,
    float* __restrict__ pout, float* __restrict__ pmax, float* __restrict__ psum)
{
    const int n    = blockIdx.x;
    const int k    = blockIdx.y;
    const int tid  = threadIdx.x;
    const int w    = tid >> 5;      // wave id 0..7
    const int lane = tid & 31;
    const int hse  = lane >> 4;     // half-select (K-pair within WMMA chunk)
    const int m    = lane & 15;     // row/col within half

    __shared__ float q_s[16 * LDSTR];   // rows 0..3 = q, rows 4..15 = 0
    __shared__ float p_s[16 * LDSTR];   // rows 0..3 = P, rows 4..15 = 0
    __shared__ float smax[8][G_];
    __shared__ float ssum[8][G_];

    const int  b    = block_groups[n];
    const int  cb   = block_list[n];
    const int  nbi  = block_indices[b];
    const int  noff = block_offsets[b];
    const bool replBlock = (cb == nbi);

    // Zero pad rows (4..15) of both A-operand arrays; stage scaled q rows.
    for (int i = tid; i < 12 * LDSTR; i += 256) {
        q_s[G_ * LDSTR + i] = 0.f;
        p_s[G_ * LDSTR + i] = 0.f;
    }
    for (int i = tid; i < G_ * D_; i += 256) {
        int g = i >> 7, d = i & (D_ - 1);
        q_s[g * LDSTR + d] = SCALE_ * query[((size_t)b * H_ + k * G_ + g) * D_ + d];
    }
    __syncthreads();

    // ---- scores: S = q (16x4 chunks) x K^T (4x16 chunks), K-dim = d ----
    const int t = 16 * w + m;   // token this lane covers as B-matrix column
    const float* kp;
    {
        bool repl = replBlock && (t == noff);
        kp = repl ? (knew + ((size_t)b * KVH_ + k) * D_)
                  : (kcache + (((size_t)cb * BS_ + t) * KVH_ + k) * D_);
    }
    v8f c = {0.f,0.f,0.f,0.f,0.f,0.f,0.f,0.f};
    #pragma unroll
    for (int kb = 0; kb < D_ / 4; kb += 8) {
        v2f av[8], bv[8];
        #pragma unroll
        for (int j = 0; j < 8; ++j) {
            int dd = 4 * (kb + j) + 2 * hse;
            av[j] = *(const v2f*)&q_s[m * LDSTR + dd];             // ds_load_b64
            bv[j] = __builtin_nontemporal_load((const v2f*)(kp + dd)); // global b64, NT
        }
        #pragma unroll
        for (int j = 0; j < 8; ++j)
            c = __builtin_amdgcn_wmma_f32_16x16x4_f32(false, av[j], false, bv[j],
                                                      (short)0, c, false, false);
    }

    const float bias = block_bias[(size_t)n * BS_ + t];

    // ---- per-g partial softmax: wave-local then cross-wave via LDS ----
    float sv[G_];
    #pragma unroll
    for (int g = 0; g < G_; ++g) {
        float v = c[g] + bias;     // valid on lanes 0..15 (M=g rows)
        sv[g] = v;
        for (int msk = 1; msk < 16; msk <<= 1)
            v = fmaxf(v, __shfl_xor(v, msk, 32));
        if (lane == 0) smax[w][g] = v;
    }
    __syncthreads();

    float bmax[G_];
    #pragma unroll
    for (int g = 0; g < G_; ++g) {
        float v = smax[0][g];
        #pragma unroll
        for (int ww = 1; ww < 8; ++ww) v = fmaxf(v, smax[ww][g]);
        bmax[g] = v;
    }

    #pragma unroll
    for (int g = 0; g < G_; ++g) {
        float p = __expf(sv[g] - bmax[g]);
        if (lane < 16) p_s[g * LDSTR + t] = p;
        float v = (lane < 16) ? p : 0.f;
        for (int msk = 1; msk < 16; msk <<= 1)
            v += __shfl_xor(v, msk, 32);
        if (lane == 0) ssum[w][g] = v;
    }
    __syncthreads();

    float bsum[G_];
    #pragma unroll
    for (int g = 0; g < G_; ++g) {
        float v = 0.f;
        #pragma unroll
        for (int ww = 0; ww < 8; ++ww) v += ssum[ww][g];
        bsum[g] = v;
    }

    // ---- O = P (16x4 chunks) x V (4x16 chunks), K-dim = tokens ----
    const int dcol = 16 * w + m;   // output d-column this lane covers
    const float* vnew_row = vnew + ((size_t)b * KVH_ + k) * D_;
    const float* vbase    = vcache + ((size_t)cb * BS_ * KVH_ + k) * D_;
    v8f acc = {0.f,0.f,0.f,0.f,0.f,0.f,0.f,0.f};
    #pragma unroll
    for (int kb = 0; kb < BS_ / 4; kb += 8) {
        v2f av[8], bv[8];
        #pragma unroll
        for (int j = 0; j < 8; ++j) {
            int s0 = 4 * (kb + j) + 2 * hse;
            av[j] = *(const v2f*)&p_s[m * LDSTR + s0];            // ds_load_b64
            const float* vp0 = (replBlock && (s0     == noff))
                             ? vnew_row : (vbase + (size_t)s0 * (KVH_ * D_));
            const float* vp1 = (replBlock && (s0 + 1 == noff))
                             ? vnew_row : (vbase + (size_t)(s0 + 1) * (KVH_ * D_));
            bv[j][0] = __builtin_nontemporal_load(vp0 + dcol);
            bv[j][1] = __builtin_nontemporal_load(vp1 + dcol);
        }
        #pragma unroll
        for (int j = 0; j < 8; ++j)
            acc = __builtin_amdgcn_wmma_f32_16x16x4_f32(false, av[j], false, bv[j],
                                                        (short)0, acc, false, false);
    }

    // ---- write per-block partials ----
    const size_t base = ((size_t)n * KVH_ + k) * G_;
    if (lane < 16) {
        #pragma unroll
        for (int g = 0; g < G_; ++g)
            pout[(base + g) * D_ + dcol] = acc[g];
    }
    if (tid == 0) {
        #pragma unroll
        for (int g = 0; g < G_; ++g) {
            pmax[base + g] = bmax[g];
            psum[base + g] = bsum[g];
        }
    }
}

// ---------------------------------------------------------------------------
// Phase 2: one workgroup per (sequence b, kv-head k); 128 threads = d index.
// Combine the sequence's per-block partials with exp-rescaling.
// ---------------------------------------------------------------------------
__global__ __launch_bounds__(128) void pa_phase2(
    const int* __restrict__ block_groups,
    const float* __restrict__ pout, const float* __restrict__ pmax,
    const float* __restrict__ psum, float* __restrict__ out)
{
    const int b = blockIdx.x;
    const int k = blockIdx.y;
    const int d = threadIdx.x;

    float gmax[G_];
    #pragma unroll
    for (int g = 0; g < G_; ++g) gmax[g] = -INFINITY;

    for (int n = 0; n < NB_; ++n) {
        if (block_groups[n] != b) continue;
        size_t base = ((size_t)n * KVH_ + k) * G_;
        #pragma unroll
        for (int g = 0; g < G_; ++g)
            gmax[g] = fmaxf(gmax[g], pmax[base + g]);
    }

    float acc[G_] = {0.f, 0.f, 0.f, 0.f};
    float gs[G_]  = {0.f, 0.f, 0.f, 0.f};
    for (int n = 0; n < NB_; ++n) {
        if (block_groups[n] != b) continue;
        size_t base = ((size_t)n * KVH_ + k) * G_;
        #pragma unroll
        for (int g = 0; g < G_; ++g) {
            float adj = __expf(pmax[base + g] - gmax[g]);
            gs[g]  += psum[base + g] * adj;
            acc[g] += pout[(base + g) * D_ + d] * adj;
        }
    }

    #pragma unroll
    for (int g = 0; g < G_; ++g)
        out[((size_t)b * H_ + k * G_ + g) * D_ + d] =
            acc[g] / fmaxf(gs[g], 1.17549435e-38f);
}

// ---------------------------------------------------------------------------
extern "C" void kernel_launch(void* const* d_in, const int* in_sizes, int n_in,
                              void* d_out, int out_size, void* d_ws, size_t ws_size,
                              hipStream_t stream) {
    const float* query         = (const float*)d_in[0];
    const float* knew          = (const float*)d_in[1];
    const float* vnew          = (const float*)d_in[2];
    const float* kcache        = (const float*)d_in[3];
    const float* vcache        = (const float*)d_in[4];
    const int*   block_list    = (const int*)d_in[5];
    const int*   block_groups  = (const int*)d_in[6];
    /* d_in[7] = block_mapping (one-hot of block_groups) — not needed */
    const float* block_bias    = (const float*)d_in[8];
    const int*   block_indices = (const int*)d_in[9];
    const int*   block_offsets = (const int*)d_in[10];
    float* out = (float*)d_out;

    // Scratch layout (~17 MB): partial outputs + per-block max/sum.
    float* pout = (float*)d_ws;
    float* pmax = pout + (size_t)NB_ * KVH_ * G_ * D_;
    float* psum = pmax + (size_t)NB_ * KVH_ * G_;

    dim3 g1(NB_, KVH_), b1(256);
    pa_phase1<<<g1, b1, 0, stream>>>(query, knew, vnew, kcache, vcache,
                                     block_list, block_groups, block_bias,
                                     block_indices, block_offsets,
                                     pout, pmax, psum);

    dim3 g2(B_, KVH_), b2(128);
    pa_phase2<<<g2, b2, 0, stream>>>(block_groups, pout, pmax, psum, out);
}